// BahdanauAttention_67456756351168
// MI455X (gfx1250) — compile-verified
//
#include <hip/hip_runtime.h>
#include <hip/hip_bf16.h>
#include <math.h>

// Problem constants (from reference): B=4, Q=256, K=512, H=512, A=512
#define B_ 4
#define Q_ 256
#define K_ 512
#define H_ 512
#define A_ 512

typedef __attribute__((ext_vector_type(2))) float v2f;
typedef __attribute__((ext_vector_type(4))) float v4f;
typedef __attribute__((ext_vector_type(8))) float v8f;

// Native CDNA5 transcendental: v_tanh_f32 (TRANS pipe, 1 instr/element).
// Called unguarded on purpose: a compile failure here is the probe signal.
__device__ __forceinline__ float fast_tanh(float x) {
    return __builtin_amdgcn_tanhf(x);
}

// ---------------------------------------------------------------------------
// Generic fp32 WMMA GEMM:  C[M,N] = X[M,Kd] @ W[Kd,N] (+ bias[N]), batched via
// blockIdx.y with element strides sxb/swb/scb. One wave computes one 16x16
// output tile via V_WMMA_F32_16X16X4_F32, stepping Kd in chunks of 4.
// Layouts per CDNA5 ISA 7.12.2:
//   A 16x4 : lane L(<16) row M=L holds K=0,1 in v0,v1 ; lanes 16-31 hold K=2,3
//   B 4x16 : v0 lanes0-15 = row K=0 (N=0..15), lanes16-31 = row K=2; v1 = K=1/K=3
//   C/D    : VGPR i -> row tileM + i + 8*(lane>=16), col tileN + (lane&15)
// ---------------------------------------------------------------------------
__global__ void wmma_gemm_f32(const float* __restrict__ Xg,
                              const float* __restrict__ Wg,
                              const float* __restrict__ bias,
                              float* __restrict__ Cg,
                              int M, int N, int Kd,
                              long long sxb, long long swb, long long scb)
{
    const float* X = Xg + (long long)blockIdx.y * sxb;
    const float* W = Wg + (long long)blockIdx.y * swb;
    float*       C = Cg + (long long)blockIdx.y * scb;

    const int lane   = threadIdx.x & 31;
    const int waveId = (blockIdx.x * blockDim.x + threadIdx.x) >> 5;
    const int tilesN = N >> 4;
    const int tileM  = (waveId / tilesN) << 4;
    const int tileN  = (waveId % tilesN) << 4;
    if (tileM >= M) return;                 // wave-uniform guard: EXEC stays full

    const int half = lane >> 4;             // 0 or 1
    const int l16  = lane & 15;
    const int arow = tileM + l16;           // A-matrix row for this lane
    const int bcol = tileN + l16;           // B-matrix col for this lane
    const int koff = half << 1;             // K sub-offset: 0 or 2

    // Hoisted induction pointers: inner loop advances by constant byte strides
    // so the compiler can fold addresses into load immediates.
    const float* xp  = X + (long long)arow * Kd + koff;        // A: K=ka, ka+1
    const float* wp0 = W + (long long)koff * N + bcol;         // B row K=ka
    const float* wp1 = wp0 + N;                                // B row K=ka+1

    v8f acc = {};
    for (int k0 = 0; k0 < Kd; k0 += 4) {
        v2f a = *(const v2f*)xp;            // 8B-aligned pair load
        v2f b;
        b.x = *wp0;
        b.y = *wp1;
        acc = __builtin_amdgcn_wmma_f32_16x16x4_f32(
                  /*neg_a=*/false, a, /*neg_b=*/false, b,
                  /*c_mod=*/(short)0, acc, /*reuse_a=*/false, /*reuse_b=*/false);
        xp  += 4;
        wp0 += 4 * (long long)N;
        wp1 += 4 * (long long)N;
    }

    const float bv    = bias ? bias[bcol] : 0.0f;
    const int   rbase = tileM + (half << 3);
#pragma unroll
    for (int i = 0; i < 8; ++i)
        C[(long long)(rbase + i) * N + bcol] = acc[i] + bv;
}

// ---------------------------------------------------------------------------
// Additive scores + masked softmax. One 256-thread block per (b,q) row.
//   scores[k] = b2 + sum_a tanh(qW[b,q,a] + kW[b,k,a]) * w2[a]   (b1 folded in qW)
// Each of the 8 waves owns k = wave, wave+8, ... ; each lane owns 4 consecutive
// 'a' values (stride 128) so all traffic is ds_load_b128 / global_load_b128.
// Then a wave shuffle reduction, masked softmax over K=512, weights written to
// the second output region.
// ---------------------------------------------------------------------------
__global__ void scores_softmax_kernel(const float* __restrict__ qW,   // [B*Q, A]
                                      const float* __restrict__ kW,   // [B, K, A]
                                      const float* __restrict__ w2,   // [A]
                                      const float* __restrict__ b2p,  // [1]
                                      const unsigned char* __restrict__ mask, // [B,Q,K] bool
                                      float* __restrict__ wout)       // [B*Q, K]
{
    __shared__ __align__(16) float qs[A_];
    __shared__ __align__(16) float w2s[A_];
    __shared__ float sc[K_];
    __shared__ float red[8];

    const int tid = threadIdx.x;            // 256 threads = 8 waves
    const int bq  = blockIdx.x;
    const int b   = bq / Q_;
    const float* qr  = qW + (long long)bq * A_;
    const float* kWb = kW + (long long)b * K_ * A_;

    for (int i = tid; i < A_; i += 256) { qs[i] = qr[i]; w2s[i] = w2[i]; }
    __syncthreads();

    const int wave = tid >> 5;
    const int lane = tid & 31;
    const float b2 = b2p[0];

    const v4f* qs4  = (const v4f*)qs;
    const v4f* w2s4 = (const v4f*)w2s;

    for (int k = wave; k < K_; k += 8) {
        const float* kr = kWb + (long long)k * A_;
        float s = 0.0f;
#pragma unroll
        for (int a0 = 0; a0 < A_; a0 += 128) {
            const int av = (a0 >> 2) + lane;            // float4 index
            v4f kv = *(const v4f*)(kr + a0 + lane * 4); // global_load_b128
            v4f qv = qs4[av];                           // ds_load_b128
            v4f wv = w2s4[av];                          // ds_load_b128
            s += fast_tanh(qv.x + kv.x) * wv.x;
            s += fast_tanh(qv.y + kv.y) * wv.y;
            s += fast_tanh(qv.z + kv.z) * wv.z;
            s += fast_tanh(qv.w + kv.w) * wv.w;
        }
#pragma unroll
        for (int off = 16; off > 0; off >>= 1)
            s += __shfl_xor(s, off, 32);
        if (lane == 0) sc[k] = s + b2;
    }
    __syncthreads();

    // Masked softmax over K=512; each thread handles 2 elements.
    const unsigned char* mrow = mask + (long long)bq * K_;
    float v0 = sc[tid];
    float v1 = sc[tid + 256];
    if (mrow[tid])       v0 = -INFINITY;
    if (mrow[tid + 256]) v1 = -INFINITY;

    // Block max
    float m = fmaxf(v0, v1);
#pragma unroll
    for (int off = 16; off > 0; off >>= 1)
        m = fmaxf(m, __shfl_xor(m, off, 32));
    if (lane == 0) red[wave] = m;
    __syncthreads();
    if (tid == 0) {
        float t = red[0];
        for (int i = 1; i < 8; ++i) t = fmaxf(t, red[i]);
        red[0] = t;
    }
    __syncthreads();
    m = red[0];
    __syncthreads();

    // Exp + block sum
    const float e0 = __expf(v0 - m);
    const float e1 = __expf(v1 - m);
    float s = e0 + e1;
#pragma unroll
    for (int off = 16; off > 0; off >>= 1)
        s += __shfl_xor(s, off, 32);
    if (lane == 0) red[wave] = s;
    __syncthreads();
    if (tid == 0) {
        float t = 0.0f;
        for (int i = 0; i < 8; ++i) t += red[i];
        red[0] = t;
    }
    __syncthreads();
    const float inv = 1.0f / red[0];

    wout[(long long)bq * K_ + tid]       = e0 * inv;
    wout[(long long)bq * K_ + tid + 256] = e1 * inv;
}

// ---------------------------------------------------------------------------
// kernel_launch
//   inputs: 0 query[B,Q,H] 1 keys[B,K,H] 2 values[B,K,H] 3 mask[B,Q,K](bool)
//           4 W1[2H,A] 5 b1[A] 6 w2[A] 7 b2[1]
//   output: context[B,Q,H] ++ weights[B,Q,K]  (fp32, concatenated flat)
//   d_ws:   qW [B*Q,A] (2MB) then kW [B,K,A] (4MB)
// ---------------------------------------------------------------------------
extern "C" void kernel_launch(void* const* d_in, const int* in_sizes, int n_in,
                              void* d_out, int out_size, void* d_ws, size_t ws_size,
                              hipStream_t stream)
{
    const float* query  = (const float*)d_in[0];
    const float* keys   = (const float*)d_in[1];
    const float* values = (const float*)d_in[2];
    const unsigned char* mask = (const unsigned char*)d_in[3];
    const float* W1 = (const float*)d_in[4];
    const float* b1 = (const float*)d_in[5];
    const float* w2 = (const float*)d_in[6];
    const float* b2 = (const float*)d_in[7];

    float* context = (float*)d_out;                          // [B,Q,H]
    float* weights = (float*)d_out + (size_t)B_ * Q_ * H_;   // [B,Q,K]

    float* qW = (float*)d_ws;                                // [B*Q, A]
    float* kW = qW + (size_t)B_ * Q_ * A_;                   // [B, K, A]

    const dim3 blk(128);                                     // 4 waves / block

    // qW = query @ W1[:H] + b1    (M=1024, N=512, Kd=512)
    {
        const int M = B_ * Q_, N = A_, Kd = H_;
        dim3 grid((M / 16) * (N / 16) / 4, 1);
        wmma_gemm_f32<<<grid, blk, 0, stream>>>(query, W1, b1, qW,
                                                M, N, Kd, 0LL, 0LL, 0LL);
    }
    // kW = keys @ W1[H:]          (M=2048, N=512, Kd=512)
    {
        const int M = B_ * K_, N = A_, Kd = H_;
        dim3 grid((M / 16) * (N / 16) / 4, 1);
        wmma_gemm_f32<<<grid, blk, 0, stream>>>(keys, W1 + (size_t)H_ * A_, nullptr, kW,
                                                M, N, Kd, 0LL, 0LL, 0LL);
    }
    // scores + masked softmax -> weights
    scores_softmax_kernel<<<dim3(B_ * Q_), dim3(256), 0, stream>>>(
        qW, kW, w2, b2, mask, weights);

    // context[b] = weights[b] @ values[b]   (per batch: 256x512x512)
    {
        const int M = Q_, N = H_, Kd = K_;
        dim3 grid((M / 16) * (N / 16) / 4, B_);
        wmma_gemm_f32<<<grid, blk, 0, stream>>>(weights, values, nullptr, context,
                                                M, N, Kd,
                                                (long long)Q_ * K_,
                                                (long long)K_ * H_,
                                                (long long)Q_ * H_);
    }
}